// SamplingBottleneckModule_72662256713982
// MI455X (gfx1250) — compile-verified
//
#include <hip/hip_runtime.h>
#include <hip/hip_bf16.h>

// ---------------------------------------------------------------------------
// SamplingBottleneck forward, CDNA5 (gfx1250, wave32).
//   1) logits_p = x @ W_probs      (WMMA f32 16x16x4, TDM-staged A slab)
//   2) logits_v = x @ W_values
//   3) row softmax (+ensure_nonzero for probs), in place
//   4) per-(b,t): 8 parallel 16-step without-replacement categorical draws,
//      discretize gathered values, scatter-mean into w_mean (LDS resident)
//   5) out = w_mean @ class_embed  (same WMMA GEMM)
// Marginals/Newton solver are dead code in the forward pass and omitted.
// ---------------------------------------------------------------------------

#define B_SZ   8
#define T_SZ   512
#define DIM_SZ 1024
#define N_SZ   1024
#define ROWS   (B_SZ * T_SZ)   // 4096
#define S_SEQ  8               // num_seqs
#define K_SEQ  16              // seq_len
#define LEVELS 128             // num_levels
#define EPS_NZ 1.2e-07f

typedef __attribute__((ext_vector_type(2))) float    v2f;
typedef __attribute__((ext_vector_type(8))) float    v8f;
typedef __attribute__((ext_vector_type(4))) unsigned u32x4;
typedef __attribute__((ext_vector_type(8))) int      i32x8;
typedef __attribute__((ext_vector_type(4))) int      i32x4;

// ------------------------------- RNG ---------------------------------------
__device__ __forceinline__ unsigned pcg_hash(unsigned v) {
  v = v * 747796405u + 2891336453u;
  unsigned w = ((v >> ((v >> 28) + 4u)) ^ v) * 277803737u;
  return (w >> 22) ^ w;
}
__device__ __forceinline__ unsigned hash2(unsigned a, unsigned b) {
  return pcg_hash(a ^ pcg_hash(b + 0x9E3779B9u));
}
__device__ __forceinline__ float u01(unsigned h) {
  return (float)(h >> 8) * (1.0f / 16777216.0f);  // [0,1)
}

// ----------------------- Tensor Data Mover (TDM) ----------------------------
// 2-D tile load Global -> LDS with the TDM pad engine inserting 1 dword of
// LDS padding every 256 dwords (bank-conflict-free pitch for WMMA A reads).
// D# packing per CDNA5 ISA. clang-23 6-arg builtin:
// (u32x4 g0, i32x8 g1, i32x4 g2, i32x4 g3, i32x8, i32 cpol).
__device__ __forceinline__ void tdm_load_2d_padded(
    unsigned lds_byte_off, const void* gptr,
    unsigned tile_w_elts, unsigned tile_h, unsigned row_stride_elts) {
  unsigned long long ga = (unsigned long long)gptr;
  u32x4 g0;
  g0[0] = 1u;                                            // count=1, user D#
  g0[1] = lds_byte_off;                                  // lds_addr
  g0[2] = (unsigned)(ga & 0xFFFFFFFFu);                  // global_addr[31:0]
  g0[3] = (unsigned)((ga >> 32) & 0x01FFFFFFu)           // global_addr[56:32]
        | (2u << 30);                                    // type=2 (image)
  i32x8 g1;
  g1[0] = (int)((2u << 16)                               // data_size = 4 B
        | (1u << 20)                                     // pad_enable
        | (7u << 22)                                     // pad_interval: 256 dw
        | (0u << 25));                                   // pad_amount: 1 dw
  g1[1] = (int)((tile_w_elts & 0xFFFFu) << 16);          // tensor_dim0[15:0]
  g1[2] = (int)(((tile_w_elts >> 16) & 0xFFFFu)          // tensor_dim0[31:16]
        | ((tile_h & 0xFFFFu) << 16));                   // tensor_dim1[15:0]
  g1[3] = (int)(((tile_h >> 16) & 0xFFFFu)               // tensor_dim1[31:16]
        | ((tile_w_elts & 0xFFFFu) << 16));              // tile_dim0
  g1[4] = (int)(tile_h & 0xFFFFu);                       // tile_dim1; tile_dim2=0
  g1[5] = (int)row_stride_elts;                          // tensor_dim0_stride lo
  g1[6] = 0;                                             // stride0 hi / stride1 lo
  g1[7] = 0;
  i32x4 z4 = {0, 0, 0, 0};                               // groups 2/3 unused (2-D)
  i32x8 z8 = {0, 0, 0, 0, 0, 0, 0, 0};
  __builtin_amdgcn_tensor_load_to_lds(g0, g1, z4, z4, z8, 0);
}

// --------------------------- WMMA f32 GEMM ---------------------------------
// C[M,N] = A[M,K] @ B[K,N], row-major fp32. 8 waves/WG; wave owns a 16x32
// stripe (2 n-tiles, A fragment reused). A slab (16xK) staged in LDS by TDM.
// Fragment layouts per CDNA5 ISA 7.12.2 (32-bit A 16x4 / C 16x16 tables).
template <int KDIM, int NDIM>
__global__ __launch_bounds__(256) void wmma_gemm_f32(
    const float* __restrict__ A, const float* __restrict__ Bm,
    float* __restrict__ C) {
  constexpr int PITCH = KDIM + (KDIM >> 8);   // TDM-padded row pitch (dwords)
  __shared__ float ldsA[16 * PITCH + 8];

  const int wave = threadIdx.x >> 5;
  const int lane = threadIdx.x & 31;
  const int hf   = lane >> 4;                 // half-wave: K offset 2*hf
  const int lm   = lane & 15;
  const int m0 = blockIdx.x * 16;
  const int n0 = (blockIdx.y * 8 + wave) * 32;

  if (wave == 0) {                            // uniform branch; TDM ignores EXEC
    tdm_load_2d_padded((unsigned)(size_t)&ldsA[0],
                       A + (size_t)m0 * KDIM, KDIM, 16, KDIM);
    __builtin_amdgcn_s_wait_tensorcnt(0);
  }
  __syncthreads();

  const float* aRow = &ldsA[lm * PITCH + 2 * hf];
  const float* B0 = Bm + (size_t)(2 * hf) * NDIM + n0 + lm;
  const float* B1 = B0 + 16;

  v8f acc0 = {}, acc1 = {};
#pragma unroll 4
  for (int k = 0; k < KDIM; k += 4) {
    const int ko = k + (k >> 8);              // skip TDM pad dwords (uniform)
    v2f a, b0, b1;
    a[0]  = aRow[ko];                         // A[m][k+2hf]     (LDS)
    a[1]  = aRow[ko + 1];                     // A[m][k+2hf+1]
    b0[0] = B0[(size_t)k * NDIM];             // B[k+2hf][n]
    b0[1] = B0[(size_t)(k + 1) * NDIM];
    b1[0] = B1[(size_t)k * NDIM];
    b1[1] = B1[(size_t)(k + 1) * NDIM];
    acc0 = __builtin_amdgcn_wmma_f32_16x16x4_f32(false, a, false, b0,
                                                 (short)0, acc0, false, false);
    acc1 = __builtin_amdgcn_wmma_f32_16x16x4_f32(false, a, false, b1,
                                                 (short)0, acc1, false, false);
  }
#pragma unroll
  for (int j = 0; j < 8; ++j) {
    const int row = j + 8 * hf;               // C/D VGPR j -> M = j + 8*half
    C[(size_t)(m0 + row) * NDIM + n0 + lm]      = acc0[j];
    C[(size_t)(m0 + row) * NDIM + n0 + 16 + lm] = acc1[j];
  }
}

// ------------------------- row softmax (in place) ---------------------------
__global__ __launch_bounds__(256) void softmax_rows(
    float* __restrict__ buf, int n, int ensure_nz) {
  __shared__ float red_max[8];
  __shared__ float red_sum[8];
  const int tid = threadIdx.x;
  const int lane = tid & 31, wave = tid >> 5;
  float* x = buf + (size_t)blockIdx.x * n;

  float4 v = *(const float4*)(x + tid * 4);

  float m = fmaxf(fmaxf(v.x, v.y), fmaxf(v.z, v.w));
#pragma unroll
  for (int o = 16; o > 0; o >>= 1) m = fmaxf(m, __shfl_xor(m, o, 32));
  if (lane == 0) red_max[wave] = m;
  __syncthreads();
  float bm = red_max[0];
#pragma unroll
  for (int i = 1; i < 8; ++i) bm = fmaxf(bm, red_max[i]);

  v.x = __expf(v.x - bm); v.y = __expf(v.y - bm);
  v.z = __expf(v.z - bm); v.w = __expf(v.w - bm);
  float s = v.x + v.y + v.z + v.w;
#pragma unroll
  for (int o = 16; o > 0; o >>= 1) s += __shfl_xor(s, o, 32);
  if (lane == 0) red_sum[wave] = s;
  __syncthreads();
  float tot = 0.f;
#pragma unroll
  for (int i = 0; i < 8; ++i) tot += red_sum[i];
  float inv = 1.0f / tot;

  v.x *= inv; v.y *= inv; v.z *= inv; v.w *= inv;
  if (ensure_nz) {
    const float sc = 1.0f - (float)n * EPS_NZ;
    v.x = v.x * sc + EPS_NZ; v.y = v.y * sc + EPS_NZ;
    v.z = v.z * sc + EPS_NZ; v.w = v.w * sc + EPS_NZ;
  }
  *(float4*)(x + tid * 4) = v;
}

// --------------- sequential sampling + discretize + scatter-mean ------------
__global__ __launch_bounds__(256) void sample_scatter(
    const float* __restrict__ probs, const float* __restrict__ values,
    float* __restrict__ wmean) {
  __shared__ float cur[S_SEQ][N_SZ];   // 32 KB: per-seq mutable prob rows
  __shared__ float acc[N_SZ];          //  4 KB: scatter accumulator
  const int row  = blockIdx.x;
  const int tid  = threadIdx.x;
  const int wave = tid >> 5;           // sequence id
  const int lane = tid & 31;
  const float* prow = probs + (size_t)row * N_SZ;
  const float* vrow = values + (size_t)row * N_SZ;

  for (int i = tid; i < N_SZ; i += 256) acc[i] = 0.f;
  for (int i = lane; i < N_SZ; i += 32) cur[wave][i] = prow[i];
  __syncthreads();

  const int base = lane * 32;
  const unsigned rseed = (unsigned)(row * S_SEQ + wave);

  for (int step = 0; step < K_SEQ; ++step) {
    float lc[32];
    float ls = 0.f;
#pragma unroll
    for (int j = 0; j < 32; ++j) { lc[j] = cur[wave][base + j]; ls += lc[j]; }

    float incl = ls;                      // inclusive scan across lanes
#pragma unroll
    for (int off = 1; off < 32; off <<= 1) {
      float y = __shfl_up(incl, off, 32);
      if (lane >= off) incl += y;
    }
    const float tot  = __shfl(incl, 31, 32);
    const float excl = incl - ls;

    const float r   = u01(hash2(0x73616D70u + step, rseed));
    const float thr = r * tot;

    int cnt = 0;                          // idx = #(inclusive cumsum < thr)
    float run = excl;
#pragma unroll
    for (int j = 0; j < 32; ++j) { run += lc[j]; cnt += (run < thr) ? 1 : 0; }
#pragma unroll
    for (int o = 16; o > 0; o >>= 1) cnt += __shfl_xor(cnt, o, 32);
    const int idx = min(cnt, N_SZ - 1);

    if (lane == 0) {                      // gather + straight-through discretize
      const float vsel = vrow[idx];
      const float r2 = u01(hash2(0x64697363u + step, rseed));
      const float di = floorf(vsel * (float)(LEVELS - 1) + 0.999f * r2);
      atomicAdd(&acc[idx], di * (1.0f / (float)(LEVELS - 1)));  // ds_add_f32
    }
    if ((idx >> 5) == lane) cur[wave][idx] = 0.f;  // without replacement
  }
  __syncthreads();

  const float invS = 1.0f / (float)S_SEQ;
  for (int i = tid; i < N_SZ; i += 256)
    wmean[(size_t)row * N_SZ + i] = acc[i] * invS;
}

// ------------------------------- launcher -----------------------------------
extern "C" void kernel_launch(void* const* d_in, const int* in_sizes, int n_in,
                              void* d_out, int out_size, void* d_ws, size_t ws_size,
                              hipStream_t stream) {
  (void)in_sizes; (void)n_in; (void)out_size; (void)ws_size;
  const float* x  = (const float*)d_in[0];   // (B,T,DIM)
  const float* Wp = (const float*)d_in[1];   // (DIM,N)
  const float* Wv = (const float*)d_in[2];   // (DIM,N)
  const float* CE = (const float*)d_in[3];   // (N,DIM)
  // d_in[4..6] = num_seqs/seq_len/num_levels scalars; fixed per reference.

  float* probs = (float*)d_ws;                       // 16 MB
  float* vals  = probs + (size_t)ROWS * N_SZ;        // 16 MB
  float* wmn   = vals  + (size_t)ROWS * N_SZ;        // 16 MB
  float* out   = (float*)d_out;                      // (B,T,DIM)

  dim3 blk(256);
  dim3 gg(ROWS / 16, N_SZ / 256);  // 16 rows/WG, 8 waves x 32 cols = 256 cols/WG

  wmma_gemm_f32<DIM_SZ, N_SZ><<<gg, blk, 0, stream>>>(x, Wp, probs);
  wmma_gemm_f32<DIM_SZ, N_SZ><<<gg, blk, 0, stream>>>(x, Wv, vals);
  softmax_rows<<<dim3(ROWS), blk, 0, stream>>>(probs, N_SZ, 1);
  softmax_rows<<<dim3(ROWS), blk, 0, stream>>>(vals,  N_SZ, 0);
  sample_scatter<<<dim3(ROWS), blk, 0, stream>>>(probs, vals, wmn);
  wmma_gemm_f32<N_SZ, DIM_SZ><<<gg, blk, 0, stream>>>(wmn, CE, out);
}